// MILCrossEntropyLoss_37769942401069
// MI455X (gfx1250) — compile-verified
//
#include <hip/hip_runtime.h>
#include <hip/hip_bf16.h>
#include <math.h>

// MIL cross-entropy loss for MI455X (gfx1250, wave32).
// HBM-streaming bound: 1.024 GB read once -> ~44us floor at 23.3 TB/s.
// One wave per bag. Data movement uses the CDNA5 async global->LDS path
// (global_load_async_to_lds_b128 / s_wait_asynccnt) with a double-buffered
// 4-row tile pipeline per wave, NT temporal hint to keep the binary-searched
// bag[] array resident in L2. Deterministic two-stage final reduction.

typedef float v4f __attribute__((ext_vector_type(4)));

#define C_DIM   128
#define ROW_B   512                 // bytes per row (128 f32)
#define T_ROWS  4                   // rows per async chunk
#define NBLOCKS 2048
#define TPB     256                 // 8 waves of 32
#define WAVES_PER_BLOCK (TPB / 32)

__device__ __forceinline__ v4f max4(v4f a, v4f b) {
    v4f r;
    r.x = fmaxf(a.x, b.x);
    r.y = fmaxf(a.y, b.y);
    r.z = fmaxf(a.z, b.z);
    r.w = fmaxf(a.w, b.w);
    return r;
}

__device__ __forceinline__ float wave_max(float v) {
    #pragma unroll
    for (int off = 16; off > 0; off >>= 1)
        v = fmaxf(v, __shfl_xor(v, off, 32));
    return v;
}

__device__ __forceinline__ float wave_sum(float v) {
    #pragma unroll
    for (int off = 16; off > 0; off >>= 1)
        v += __shfl_xor(v, off, 32);
    return v;
}

// lower_bound over sorted int array (wave-uniform; L2-resident thanks to NT
// streaming of the big input)
__device__ __forceinline__ int lower_bound(const int* __restrict__ a, int lo, int hi, int key) {
    while (lo < hi) {
        int mid = (lo + hi) >> 1;
        if (a[mid] < key) lo = mid + 1; else hi = mid;
    }
    return lo;
}

// CDNA5 async copy: 1 row = 32 lanes x 16B global -> LDS. GVS mode:
// saddr = 64-bit base SGPR pair, vaddr = per-lane u32 byte offset,
// vdst = per-lane u32 LDS byte address. Tracked by ASYNCcnt.
__device__ __forceinline__ void async_row_to_lds(const float* base, unsigned goff, unsigned ldsoff) {
    asm volatile("global_load_async_to_lds_b128 %0, %1, %2 th:TH_LOAD_NT"
                 :: "v"(ldsoff), "v"(goff), "s"(base)
                 : "memory");
}

__global__ __launch_bounds__(TPB) void mil_ce_bag_kernel(
    const float* __restrict__ input,   // [N, 128]
    const int*   __restrict__ target,  // [M]
    const int*   __restrict__ bag,     // [N], sorted contiguous 0..M-1
    float*       __restrict__ partials,// [nblocks]
    int N, int M)
{
    // per-wave double buffer: 2 tiles x T_ROWS x 512B  (32 KB / block)
    __shared__ float s_tile[WAVES_PER_BLOCK][2][T_ROWS][C_DIM];
    __shared__ float s_wave[WAVES_PER_BLOCK];

    const int lane   = threadIdx.x & 31;
    const int wslot  = threadIdx.x >> 5;
    const int wave   = blockIdx.x * WAVES_PER_BLOCK + wslot;
    const int nwaves = gridDim.x * WAVES_PER_BLOCK;

    // LDS byte address of this wave's buffer 0 (flat->LDS uses addr[31:0])
    const unsigned lbase   = (unsigned)(size_t)&s_tile[wslot][0][0][0];
    const unsigned lane16  = (unsigned)(lane << 4);
    const unsigned BUF_B   = T_ROWS * ROW_B;

    float acc = 0.0f;

    for (int b = wave; b < M; b += nwaves) {
        // --- bag boundaries via wave-uniform binary search ---
        const int start = lower_bound(bag, 0, N, b);
        __builtin_prefetch(input + (size_t)start * C_DIM + lane * 4, 0, 1);
        const int end   = lower_bound(bag, start, N, b + 1);
        const int nrows = end - start;
        const int nchunks = (nrows + T_ROWS - 1) / T_ROWS;

        // --- prologue: issue chunk 0 (padded to T_ROWS async ops) ---
        #pragma unroll
        for (int r = 0; r < T_ROWS; ++r) {
            int rr = r < nrows ? r : nrows - 1;
            unsigned go = (unsigned)((size_t)(start + rr) * ROW_B) + lane16;
            async_row_to_lds(input, go, lbase + (unsigned)(r * ROW_B) + lane16);
        }

        v4f m;
        m.x = m.y = m.z = m.w = -INFINITY;
        int cur = 0;

        for (int c = 0; c < nchunks; ++c) {
            if (c + 1 < nchunks) {
                // issue next chunk into the other buffer (padded to T_ROWS)
                const int rb = (c + 1) * T_ROWS;
                #pragma unroll
                for (int r = 0; r < T_ROWS; ++r) {
                    int rr = (rb + r) < nrows ? (rb + r) : nrows - 1;
                    unsigned go = (unsigned)((size_t)(start + rr) * ROW_B) + lane16;
                    async_row_to_lds(input, go,
                        lbase + (unsigned)((cur ^ 1) * BUF_B + r * ROW_B) + lane16);
                }
                // <=T_ROWS outstanding (in-order completion) => chunk c landed
                asm volatile("s_wait_asynccnt 4" ::: "memory");
            } else {
                asm volatile("s_wait_asynccnt 0" ::: "memory");
            }

            // --- consume chunk c from LDS: lane owns classes [4L,4L+3] ---
            const int rows_here = (nrows - c * T_ROWS) < T_ROWS ? (nrows - c * T_ROWS) : T_ROWS;
            for (int r = 0; r < rows_here; ++r) {
                v4f v = *(const v4f*)&s_tile[wslot][cur][r][lane * 4];
                m = max4(m, v);
            }
            cur ^= 1;
        }

        // --- logsumexp over the 128 bag-level maxima ---
        float lm = fmaxf(fmaxf(m.x, m.y), fmaxf(m.z, m.w));
        const float gmax = wave_max(lm);

        float es = expf(m.x - gmax) + expf(m.y - gmax)
                 + expf(m.z - gmax) + expf(m.w - gmax);
        const float esum = wave_sum(es);
        const float logz = gmax + logf(esum);

        // --- picked logit: class t lives in lane t>>2, component t&3 ---
        const int t = target[b];
        const int cc = t & 3;
        float sel = (cc == 0) ? m.x : (cc == 1) ? m.y : (cc == 2) ? m.z : m.w;
        const float picked = __shfl(sel, t >> 2, 32);

        acc += logz - picked;
    }

    // --- block partial (every block writes unconditionally) ---
    if (lane == 0) s_wave[wslot] = acc;
    __syncthreads();
    if (threadIdx.x == 0) {
        float s = 0.0f;
        #pragma unroll
        for (int i = 0; i < WAVES_PER_BLOCK; ++i) s += s_wave[i];
        partials[blockIdx.x] = s;
    }
}

__global__ __launch_bounds__(TPB) void mil_ce_finalize_kernel(
    const float* __restrict__ partials, int nparts,
    float* __restrict__ out, float inv_m)
{
    __shared__ float sh[TPB];
    float s = 0.0f;
    for (int i = threadIdx.x; i < nparts; i += TPB) s += partials[i];
    sh[threadIdx.x] = s;
    __syncthreads();
    #pragma unroll
    for (int st = TPB / 2; st > 0; st >>= 1) {
        if (threadIdx.x < st) sh[threadIdx.x] += sh[threadIdx.x + st];
        __syncthreads();
    }
    if (threadIdx.x == 0) out[0] = sh[0] * inv_m;
}

extern "C" void kernel_launch(void* const* d_in, const int* in_sizes, int n_in,
                              void* d_out, int out_size, void* d_ws, size_t ws_size,
                              hipStream_t stream) {
    const float* input  = (const float*)d_in[0];   // [N, 128] f32
    const int*   target = (const int*)d_in[1];     // [M] i32
    const int*   bag    = (const int*)d_in[2];     // [N] i32 sorted

    const int N = in_sizes[0] / C_DIM;
    const int M = in_sizes[1];

    float* partials = (float*)d_ws;
    int nblocks = NBLOCKS;
    const size_t ws_floats = ws_size / sizeof(float);
    if ((size_t)nblocks > ws_floats) nblocks = (int)ws_floats;
    if (nblocks < 1) nblocks = 1;

    mil_ce_bag_kernel<<<nblocks, TPB, 0, stream>>>(
        input, target, bag, partials, N, M);

    mil_ce_finalize_kernel<<<1, TPB, 0, stream>>>(
        partials, nblocks, (float*)d_out, 1.0f / (float)M);
}